// NashCascadeNetwork_36893769073093
// MI455X (gfx1250) — compile-verified
//
#include <hip/hip_runtime.h>
#include <math.h>

// ---- problem geometry (from reference) ----
#define NBQ            4194304               // total buckets (8 layers x 524288)
#define NSP            8                     // spigots per bucket
#define BPL            524288                // buckets in last layer
#define BLOCK          256
#define GRID           (NBQ / BLOCK)         // 16384 blocks
#define FIRST_LAST_BLK ((NBQ - BPL) / BLOCK) // 14336
#define NPART          (BPL / BLOCK)         // 2048 partial sums
#define TWO_G          19.6f                 // 2 * 9.8

typedef __attribute__((ext_vector_type(4))) unsigned int u32x4;
typedef __attribute__((ext_vector_type(8))) int          i32x8;
typedef __attribute__((ext_vector_type(4))) int          i32x4;

#if __has_builtin(__builtin_amdgcn_tensor_load_to_lds)
#define HAS_TDM 1
#else
#define HAS_TDM 0
#endif

// ---- fast transcendental paths (gfx1250 hardware trans ops) ----
__device__ __forceinline__ float fast_tanh(float x)
{
#if __has_builtin(__builtin_amdgcn_tanhf)
    return __builtin_amdgcn_tanhf(x);     // v_tanh_f32 (TRANS op)
#else
    return tanhf(x);
#endif
}

__device__ __forceinline__ float fast_sqrt(float x)
{
#if __has_builtin(__builtin_amdgcn_sqrtf)
    return __builtin_amdgcn_sqrtf(x);     // raw v_sqrt_f32, no fixup sequence
#else
    return sqrtf(x);
#endif
}

// q = theta * sqrt(2*g*h) * 0.5*(tanh(h)+1) * area,  h = max(0, H - height)
__device__ __forceinline__ float spigot_flow(float Hb, float sh, float sa, float th)
{
    float h   = fmaxf(0.0f, Hb - sh);
    float mod = __builtin_fmaf(0.5f, fast_tanh(h), 0.5f);
    return th * fast_sqrt(TWO_G * h) * mod * sa;
}

// ---- Tensor Data Mover: DMA one contiguous row of nDw dwords global -> LDS ----
// D# packing per CDNA5 ISA (groups 0/1; trailing groups
// zero; tile is degenerate 2D: tile_dim0 = nDw dwords, tile_dim1 = 1).
// Toolchain exposes the 6-arg builtin:
//   (uint32x4 g0, int32x8 g1, int32x4, int32x4, int32x8, i32 cpol)
__device__ __forceinline__ void tdm_load_row(unsigned lds_off, const void* gsrc, unsigned nDw)
{
#if HAS_TDM
    unsigned long long ga = (unsigned long long)gsrc;
    u32x4 g0 = { 1u,                                              // count=1, user descriptor
                 lds_off,                                         // lds_addr [63:32]
                 (unsigned)ga,                                    // global_addr [95:64]
                 ((unsigned)(ga >> 32) & 0x01FFFFFFu) | (2u << 30) }; // addr[120:96] | type=2
    i32x8 g1 = { (int)0x00020000u,                 // wg_mask=0, data_size=2 (4 bytes)
                 (int)((nDw & 0xFFFFu) << 16),     // tensor_dim0[15:0] at bits[63:48]
                 (int)((nDw >> 16) | (1u << 16)),  // tensor_dim0[31:16] | tensor_dim1=1
                 (int)(nDw << 16),                 // tile_dim0 at bits[127:112] (nDw <= 65535)
                 1,                                // tile_dim1=1, tile_dim2=0 (unused)
                 (int)nDw,                         // tensor_dim0_stride low 32
                 0, 0 };
    i32x4 z4 = { 0, 0, 0, 0 };
    i32x8 z8 = { 0, 0, 0, 0, 0, 0, 0, 0 };
    __builtin_amdgcn_tensor_load_to_lds(g0, g1, z4, z4, z8, 0);
#else
    (void)lds_off; (void)gsrc; (void)nDw;
#endif
}

__device__ __forceinline__ void wait_tensorcnt0()
{
#if __has_builtin(__builtin_amdgcn_s_wait_tensorcnt)
    __builtin_amdgcn_s_wait_tensorcnt(0);
#else
    asm volatile("s_wait_tensorcnt 0x0" ::: "memory");
#endif
}

// ---- fused main pass: s_q, outflow shift, H_new, last-layer block partials ----
__global__ void __launch_bounds__(BLOCK) nash_main(
    const float* __restrict__ H, const float* __restrict__ S,
    const float* __restrict__ theta, const float* __restrict__ precip,
    float* __restrict__ Hnew, float* __restrict__ sq, float* __restrict__ partials)
{
    __shared__ float ldsTheta[BLOCK * NSP];      // 8 KB tile
    __shared__ float ldsS[BLOCK * NSP * 2];      // 16 KB tile
    __shared__ float ldsOut[BLOCK];              // per-bucket outflow

    const int      tid = threadIdx.x;
    const unsigned blk = blockIdx.x;
    const size_t   b   = (size_t)blk * BLOCK + tid;

    const size_t thetaBase = (size_t)blk * BLOCK * NSP;
    const size_t sBase     = (size_t)blk * BLOCK * NSP * 2;

#if HAS_TDM
    // One wave per block issues the two TDM descriptors (EXEC is ignored by
    // tensor ops); it waits TENSORcnt==0, then the barrier publishes the tiles.
    if (tid < 32) {
        tdm_load_row((unsigned)(size_t)&ldsTheta[0], theta + thetaBase, BLOCK * NSP);
        tdm_load_row((unsigned)(size_t)&ldsS[0],     S     + sBase,     BLOCK * NSP * 2);
        wait_tensorcnt0();
    }
#else
    {   // cooperative vectorized staging fallback
        const float4* gt = (const float4*)(theta + thetaBase);
        float4*       lt = (float4*)ldsTheta;
        for (int i = tid; i < BLOCK * NSP / 4; i += BLOCK) lt[i] = gt[i];
        const float4* gs = (const float4*)(S + sBase);
        float4*       ls = (float4*)ldsS;
        for (int i = tid; i < BLOCK * NSP * 2 / 4; i += BLOCK) ls[i] = gs[i];
    }
#endif
    __syncthreads();

    const float Hb = H[b];
    float q[NSP];
    float outflow = 0.0f;
#pragma unroll
    for (int j = 0; j < NSP; ++j) {
        float val = spigot_flow(Hb,
                                ldsS[tid * (NSP * 2) + 2 * j],
                                ldsS[tid * (NSP * 2) + 2 * j + 1],
                                ldsTheta[tid * NSP + j]);
        q[j] = val;
        outflow += val;
    }

    // s_q: 32 contiguous bytes per thread -> two B128 stores, 1 KB/wave contiguous
    float4* sqv = (float4*)(sq + b * NSP);
    sqv[0] = make_float4(q[0], q[1], q[2], q[3]);
    sqv[1] = make_float4(q[4], q[5], q[6], q[7]);

    ldsOut[tid] = outflow;
    __syncthreads();

    // shift: inflow[b] = outflow[b-1] (precip for b==0); block-edge bucket is
    // recomputed from global memory (1/256 overhead, avoids a second pass)
    float inflow;
    if (tid == 0) {
        if (b == 0) {
            inflow = precip[0];
        } else {
            const size_t p  = b - 1;
            const float  Hp = H[p];
            float acc = 0.0f;
#pragma unroll
            for (int j = 0; j < NSP; ++j)
                acc += spigot_flow(Hp,
                                   S[p * (NSP * 2) + 2 * j],
                                   S[p * (NSP * 2) + 2 * j + 1],
                                   theta[p * NSP + j]);
            inflow = acc;
        }
    } else {
        inflow = ldsOut[tid - 1];
    }
    Hnew[b] = Hb + inflow - outflow;

    // deterministic last-layer partial: fixed-order LDS tree reduction
    if (blk >= FIRST_LAST_BLK) {
        __syncthreads();
        for (int off = BLOCK / 2; off > 0; off >>= 1) {
            if (tid < off) ldsOut[tid] += ldsOut[tid + off];
            __syncthreads();
        }
        if (tid == 0) partials[blk - FIRST_LAST_BLK] = ldsOut[0];
    }
}

// ---- deterministic final reduction of the 2048 block partials ----
__global__ void __launch_bounds__(BLOCK) nash_tail_reduce(
    const float* __restrict__ partials, float* __restrict__ out_scalar)
{
    __shared__ float red[BLOCK];
    const int tid = threadIdx.x;
    float s = 0.0f;
#pragma unroll
    for (int k = 0; k < NPART / BLOCK; ++k) s += partials[tid + k * BLOCK];
    red[tid] = s;
    __syncthreads();
    for (int off = BLOCK / 2; off > 0; off >>= 1) {
        if (tid < off) red[tid] += red[tid + off];
        __syncthreads();
    }
    if (tid == 0) out_scalar[0] = red[0];   // full overwrite every call
}

extern "C" void kernel_launch(void* const* d_in, const int* in_sizes, int n_in,
                              void* d_out, int out_size, void* d_ws, size_t ws_size,
                              hipStream_t stream)
{
    (void)in_sizes; (void)n_in; (void)out_size; (void)ws_size;

    const float* H      = (const float*)d_in[0];   // [NBQ]
    const float* S      = (const float*)d_in[1];   // [NBQ, NSP, 2]
    const float* theta  = (const float*)d_in[2];   // [NBQ * NSP]
    const float* precip = (const float*)d_in[3];   // scalar

    float* out      = (float*)d_out;
    float* Hnew     = out;                                 // [NBQ]
    float* sq       = out + NBQ;                           // [NBQ * NSP]
    float* scalar   = out + NBQ + (size_t)NBQ * NSP;       // [1]
    float* partials = (float*)d_ws;                        // [NPART] scratch

    nash_main<<<GRID, BLOCK, 0, stream>>>(H, S, theta, precip, Hnew, sq, partials);
    nash_tail_reduce<<<1, BLOCK, 0, stream>>>(partials, scalar);
}